// GenerativeUpsample_790273983061
// MI455X (gfx1250) — compile-verified
//
#include <hip/hip_runtime.h>
#include <hip/hip_bf16.h>

typedef __attribute__((ext_vector_type(2))) float v2f;
typedef __attribute__((ext_vector_type(8))) float v8f;

#define N_PTS 16384
#define M_PTS 16384
#define C_IN  64
#define C_UP  64
#define KTH   8192
#define BB    128            // batch separation scale; BB^2=16384 > max in-batch d2 (11907)
#define DIST_UB 4.0f
#define BM_WORDS ((4*64*64*64)/32)   // 2^20 bits = 32768 u32

// workspace layout (bytes)
#define OFF_BM   256
#define OFF_TB0  131584                     // 256 + 128KB bitmap + 256 pad
#define OFF_TB1  (OFF_TB0 + M_PTS*8)
#define OFF_TSQ  (OFF_TB1 + M_PTS*8)

// ---- order-preserving float<->uint key maps (for k-th smallest selection) ----
static __device__ __forceinline__ unsigned fkey(float f) {
  unsigned b = __float_as_uint(f);
  return (b & 0x80000000u) ? ~b : (b | 0x80000000u);
}
static __device__ __forceinline__ float finv(unsigned u) {
  return (u & 0x80000000u) ? __uint_as_float(u ^ 0x80000000u) : __uint_as_float(~u);
}

// ---- K0: zero the coordinate-key bitmap ----
__global__ void k_zero(unsigned* __restrict__ bm, int n) {
  int i = blockIdx.x * blockDim.x + threadIdx.x;
  if (i < n) bm[i] = 0u;
}

// ---- K1: scatter target keys into bitmap + precompute WMMA-B-layout target data ----
__global__ void k_prep(const int* __restrict__ tc, unsigned* __restrict__ bm,
                       v2f* __restrict__ tB0, v2f* __restrict__ tB1,
                       float* __restrict__ tsq) {
  int i = blockIdx.x * blockDim.x + threadIdx.x;
  if (i < M_PTS) {
    const int4 t = ((const int4*)tc)[i];         // (batch, x, y, z)
    unsigned key = (unsigned)((((t.x*64 + t.y)*64 + t.z)*64) + t.w);
    atomicOr(&bm[key >> 5], 1u << (key & 31));
    int tb = t.x * BB;
    v2f b0; b0.x = (float)t.y; b0.y = (float)t.z;          // K components 0,1 (x,y)
    v2f b1; b1.x = (float)t.w; b1.y = (float)tb;           // K components 2,3 (z, batch*BB)
    tB0[i] = b0;
    tB1[i] = b1;
    tsq[i] = (float)(t.y*t.y + t.z*t.z + t.w*t.w + tb*tb); // exact: < 2^24
  }
}

// ---- K2: fea_up = relu(fea @ W_up); pred = fea_up @ W_cls ----
// one wave per 16-row tile, 4 col-tiles of 16, K-loop of 16 WMMA_F32_16X16X4 steps.
// W_up staged in LDS pre-paired (rows kk,kk+1 per column) so each B operand is a
// single aligned ds_load_b64 into an even VGPR pair (no register shuffling).
__global__ __launch_bounds__(256) void k_gemm(const float* __restrict__ fea,
                                              const float* __restrict__ Wup,
                                              const float* __restrict__ Wcls,
                                              float* __restrict__ pred,
                                              float* __restrict__ feaup) {
  __shared__ v2f   sWp[(C_IN/2) * C_UP];   // sWp[kk2*64 + c] = (W[2kk2][c], W[2kk2+1][c])
  __shared__ float sWc[C_UP];
  int tid = threadIdx.x;
  for (int idx = tid; idx < (C_IN/2) * C_UP; idx += 256) {
    int kk2 = idx >> 6, c = idx & 63;
    v2f w; w.x = Wup[(2*kk2) * C_UP + c]; w.y = Wup[(2*kk2 + 1) * C_UP + c];
    sWp[idx] = w;
  }
  if (tid < C_UP) sWc[tid] = Wcls[tid];
  __syncthreads();

  int wave = tid >> 5, lane = tid & 31;
  int half = lane >> 4, m = lane & 15;
  int r0 = (blockIdx.x * 8 + wave) * 16;

  v8f acc[4] = {{0,0,0,0,0,0,0,0},{0,0,0,0,0,0,0,0},{0,0,0,0,0,0,0,0},{0,0,0,0,0,0,0,0}};
  const float* arow = fea + (size_t)(r0 + m) * C_IN + 2 * half;  // A: lanes 0-15 K={0,1}, 16-31 K={2,3}

  #pragma unroll
  for (int k = 0; k < C_IN; k += 4) {
    v2f a = *(const v2f*)(arow + k);                       // 8B aligned (k+2*half even)
    const v2f* wk = sWp + (k/2 + half) * C_UP + m;         // row pair (k+2h, k+2h+1)
    #pragma unroll
    for (int t = 0; t < 4; ++t) {
      v2f b = wk[t * 16];                                  // one ds_load_b64
      acc[t] = __builtin_amdgcn_wmma_f32_16x16x4_f32(false, a, false, b,
                                                     (short)0, acc[t], false, false);
    }
  }

  // relu + store fea_up + accumulate pred dot-product partials
  float part[8];
  #pragma unroll
  for (int i = 0; i < 8; ++i) part[i] = 0.0f;
  #pragma unroll
  for (int t = 0; t < 4; ++t) {
    float wc = sWc[t * 16 + m];
    #pragma unroll
    for (int i = 0; i < 8; ++i) {
      float v = acc[t][i];
      v = v > 0.0f ? v : 0.0f;                             // relu
      feaup[(size_t)(r0 + i + 8 * half) * C_UP + t * 16 + m] = v;
      part[i] += v * wc;
    }
  }
  #pragma unroll
  for (int i = 0; i < 8; ++i) {
    #pragma unroll
    for (int mask = 1; mask < 16; mask <<= 1)
      part[i] += __shfl_xor(part[i], mask, 32);
  }
  if (m == 0) {
    #pragma unroll
    for (int i = 0; i < 8; ++i) pred[r0 + i + 8 * half] = part[i];
  }
}

// ---- K3: 8192-th smallest of pred via 32-step radix bisection on uint keys ----
__global__ __launch_bounds__(1024) void k_select(const float* __restrict__ pred,
                                                 float* __restrict__ thres) {
  extern __shared__ unsigned sk[];     // 16384 keys = 64 KB LDS
  __shared__ int scnt;
  int tid = threadIdx.x;
  for (int i = tid; i < N_PTS; i += 1024) sk[i] = fkey(pred[i]);
  unsigned prefix = 0;
  for (int b = 31; b >= 0; --b) {
    if (tid == 0) scnt = 0;
    __syncthreads();
    unsigned test = prefix | (1u << b);
    int c = 0;
    for (int i = tid; i < N_PTS; i += 1024) c += (sk[i] < test) ? 1 : 0;
    atomicAdd(&scnt, c);
    __syncthreads();
    int total = scnt;
    if (total < KTH) prefix = test;    // k-th smallest >= test
    __syncthreads();                   // protect scnt reuse next iteration
  }
  if (tid == 0) thres[0] = finv(prefix);
}

// ---- K4: brute-force nearest-target scan via WMMA_F32_16X16X4 + finalize ----
// d2 = |t4|^2 - 2 p4.t4 + |p4|^2, p4/t4 = (x,y,z,batch*BB); all-int, f32-exact.
__global__ __launch_bounds__(256) void k_final(const int* __restrict__ coords,
                                               const v2f* __restrict__ tB0,
                                               const v2f* __restrict__ tB1,
                                               const float* __restrict__ tsq,
                                               const float* __restrict__ pred,
                                               const float* __restrict__ thresp,
                                               const unsigned* __restrict__ bm,
                                               float* __restrict__ fea_io,
                                               float* __restrict__ keep_o,
                                               float* __restrict__ kt_o,
                                               float* __restrict__ loss_o) {
  __shared__ float smin[8][16];
  __shared__ float skeep[8][16];
  int tid = threadIdx.x;
  int wave = tid >> 5, lane = tid & 31;
  int half = lane >> 4, m = lane & 15;
  int r0 = (blockIdx.x * 8 + wave) * 16;

  // this lane serves A-matrix row m (and, for lanes<16, the final per-row logic of row m)
  const int4 pcd = ((const int4*)coords)[r0 + m];
  int pb = pcd.x, px = pcd.y, py = pcd.z, pz = pcd.w;
  float pc4[4] = {(float)px, (float)py, (float)pz, (float)(pb * BB)};
  v2f a; a.x = -2.0f * pc4[2*half]; a.y = -2.0f * pc4[2*half+1];
  float psq = (float)(px*px + py*py + pz*pz + pb*pb*BB*BB);

  v8f mn;
  #pragma unroll
  for (int i = 0; i < 8; ++i) mn[i] = 3.0e38f;

  const v2f* tB = half ? tB1 : tB0;         // this lane's 2 K-components, ~L2-resident
  #pragma unroll 4
  for (int n0 = 0; n0 < M_PTS; n0 += 16) {
    v2f  b  = tB[n0 + m];                   // one global_load_b64
    float ts = tsq[n0 + m];                 // one global_load_b32
    v8f c;
    #pragma unroll
    for (int i = 0; i < 8; ++i) c[i] = ts;  // C preloaded with per-column |t4|^2
    v8f d = __builtin_amdgcn_wmma_f32_16x16x4_f32(false, a, false, b,
                                                  (short)0, c, false, false);
    #pragma unroll
    for (int i = 0; i < 8; ++i) mn[i] = fminf(mn[i], d[i]);
  }

  // min over the 16 target columns held across each 16-lane half
  #pragma unroll
  for (int i = 0; i < 8; ++i) {
    #pragma unroll
    for (int mask = 1; mask < 16; mask <<= 1)
      mn[i] = fminf(mn[i], __shfl_xor(mn[i], mask, 32));
  }
  if (m == 0) {
    #pragma unroll
    for (int i = 0; i < 8; ++i) smin[wave][i + 8 * half] = mn[i];
  }
  __syncthreads();

  if (lane < 16) {                       // half==0, so pcd is row m's coord
    int row = r0 + m;
    float d2 = fmaxf(smin[wave][m] + psq, 0.0f);   // cross-batch mins >= BB^2 >> 4 ("inf")
    float p = pred[row];
    bool pmk = p  > DIST_UB;
    bool tmk = d2 > DIST_UB;
    float loss = tmk ? (pmk ? p : DIST_UB) : d2;
    unsigned key = (unsigned)((((pb*64 + px)*64 + py)*64) + pz);
    bool kt = (bm[key >> 5] >> (key & 31)) & 1u;
    bool keep = (p <= thresp[0]) || kt;
    keep_o[row] = keep ? 1.0f : 0.0f;
    kt_o[row]   = kt   ? 1.0f : 0.0f;
    loss_o[row] = loss;
    skeep[wave][m] = keep ? 1.0f : 0.0f;
  }
  __syncthreads();

  // fea_pruned = fea_up * keep, in place on d_out's fea region (rewritten by K2 each call)
  #pragma unroll 4
  for (int r = 0; r < 16; ++r) {
    float kf = skeep[wave][r];
    float* rowp = fea_io + (size_t)(r0 + r) * C_UP;
    rowp[lane]      *= kf;
    rowp[lane + 32] *= kf;
  }
}

extern "C" void kernel_launch(void* const* d_in, const int* in_sizes, int n_in,
                              void* d_out, int out_size, void* d_ws, size_t ws_size,
                              hipStream_t stream) {
  const float* fea     = (const float*)d_in[0];
  const float* Wup     = (const float*)d_in[1];
  const float* Wcls    = (const float*)d_in[2];
  const int*   coords  = (const int*)d_in[3];
  const int*   tcoords = (const int*)d_in[4];

  float* out    = (float*)d_out;
  float* pred   = out;                       // [N]
  float* feaup  = out + N_PTS;               // [N*64] fea_pruned region (built in place)
  float* keep_o = out + (size_t)65 * N_PTS;  // [N]
  float* kt_o   = out + (size_t)66 * N_PTS;  // [N]
  float* loss_o = out + (size_t)67 * N_PTS;  // [N]

  char* ws = (char*)d_ws;
  float*    thres = (float*)ws;
  unsigned* bm    = (unsigned*)(ws + OFF_BM);
  v2f*      tB0   = (v2f*)(ws + OFF_TB0);
  v2f*      tB1   = (v2f*)(ws + OFF_TB1);
  float*    tsq   = (float*)(ws + OFF_TSQ);

  k_zero  <<<(BM_WORDS + 255) / 256, 256, 0, stream>>>(bm, BM_WORDS);
  k_prep  <<<(M_PTS + 255) / 256,    256, 0, stream>>>(tcoords, bm, tB0, tB1, tsq);
  k_gemm  <<<128, 256, 0, stream>>>(fea, Wup, Wcls, pred, feaup);
  k_select<<<1, 1024, N_PTS * sizeof(unsigned), stream>>>(pred, thres);
  k_final <<<128, 256, 0, stream>>>(coords, tB0, tB1, tsq, pred, thres, bm,
                                    feaup, keep_o, kt_o, loss_o);
}